// DEFT_func_83751862272557
// MI455X (gfx1250) — compile-verified
//
#include <hip/hip_runtime.h>

#define B_TOT 24576
#define NV    129
#define NE    128
#define BLOCK 128                    // 4 waves per block
#define WAVES 4
#define TILE  16                     // edges per staged tile
#define NTILES (NE / TILE)           // 8
#define ROWF  ((TILE + 1) * 3)       // 51 floats staged per rod per tile (odd -> LDS conflict-free)
#define RPW   32                     // rods per wave
#define WBUF  (RPW * ROWF)           // floats per wave per buffer = 1632
#define BUFF  (WAVES * WBUF)         // floats per buffer = 6528  (x2 buffers = 52,224 B LDS)

typedef unsigned int       u32;
typedef unsigned long long u64;
typedef u32 u32x4 __attribute__((ext_vector_type(4)));
typedef int i32x8 __attribute__((ext_vector_type(8)));
typedef int i32x4 __attribute__((ext_vector_type(4)));

// Issue one TDM 2-D tile load: 32 rows of ROWF floats, row stride 387 floats,
// from global `gaddr` into LDS byte offset `lds_byte`. D# packed per ISA 08 §8.
__device__ __forceinline__ void tdm_issue(u32 lds_byte, u64 gaddr)
{
    u32x4 g0;
    g0.x = 1u;                                   // count=1, user descriptor, no gather
    g0.y = lds_byte;                             // lds_addr (bytes)
    g0.z = (u32)gaddr;                           // global_addr[31:0]
    g0.w = (u32)(gaddr >> 32) | (2u << 30);      // global_addr[56:32] | type=2 ("image")
    i32x8 g1;
    g1[0] = (int)(2u << 16);                     // data_size=4B; wg_mask=0; no flags
    g1[1] = (int)((u32)ROWF << 16);              // tensor_dim0[15:0] @bits63:48
    g1[2] = (int)((u32)RPW  << 16);              // tensor_dim0 hi=0 | tensor_dim1[15:0]=32
    g1[3] = (int)((u32)ROWF << 16);              // tensor_dim1 hi=0 | tile_dim0=ROWF
    g1[4] = (int)RPW;                            // tile_dim1=32 | tile_dim2=0
    g1[5] = (int)(NV * 3);                       // tensor_dim0_stride = 387 elements
    g1[6] = 0;                                   // stride hi | tensor_dim1_stride lo
    g1[7] = 0;
    i32x4 z4 = {0, 0, 0, 0};                     // groups 2/3 unused (2-D tensor)
    i32x8 z8 = {0, 0, 0, 0, 0, 0, 0, 0};         // extra arg in 6-arg (clang-23) form
    __builtin_amdgcn_tensor_load_to_lds(g0, g1, z4, z4, z8, 0);
}

__global__ __launch_bounds__(BLOCK) void deft_rod_frames(
    const float* __restrict__ verts,       // (B,129,3)
    const float* __restrict__ init_direct, // (B,3)
    const float* __restrict__ m_theta,     // (B,128)
    const float* __restrict__ edge_mask,   // (B,128,1)
    const float* __restrict__ restEdgeL,   // (B,128)
    float* __restrict__ out)               // (B,128,5,3)
{
    __shared__ float lds[2 * BUFF];

    const int tid  = (int)threadIdx.x;
    const int lane = tid & 31;
    const int w    = __builtin_amdgcn_readfirstlane(tid >> 5);
    const int rod  = (int)blockIdx.x * BLOCK + tid;

    const u32 ldsBase = (u32)(u64)&lds[0];                       // flat LDS addr low 32 = LDS offset
    const u64 gWave   = (u64)verts +
                        (u64)((int)blockIdx.x * BLOCK + w * RPW) * (u64)(NV * 3 * 4);

    // ---- prime the pipeline: tile 0 -> buffer 0, per wave ----
    tdm_issue(ldsBase + (u32)(w * WBUF * 4), gWave);
    __builtin_amdgcn_s_wait_tensorcnt(0);

    // ---- u0 = normalize(cross(cross(e0, d), e0)) ----
    float ux, uy, uz;
    {
        const float* r0 = &lds[w * WBUF + lane * ROWF];
        const float ax = r0[0], ay = r0[1], az = r0[2];
        const float e0x = r0[3] - ax, e0y = r0[4] - ay, e0z = r0[5] - az;
        const float dx = init_direct[(size_t)rod * 3 + 0];
        const float dy = init_direct[(size_t)rod * 3 + 1];
        const float dz = init_direct[(size_t)rod * 3 + 2];
        const float nx = e0y * dz - e0z * dy;
        const float ny = e0z * dx - e0x * dz;
        const float nz = e0x * dy - e0y * dx;
        const float tx = ny * e0z - nz * e0y;
        const float ty = nz * e0x - nx * e0z;
        const float tz = nx * e0y - ny * e0x;
        const float inv = 1.0f / fmaxf(sqrtf(tx * tx + ty * ty + tz * tz), 1e-12f);
        ux = tx * inv; uy = ty * inv; uz = tz * inv;
    }

    const float* rl_g  = restEdgeL + (size_t)rod * NE;
    const float* mk_g  = edge_mask + (size_t)rod * NE;
    const float* th_g  = m_theta   + (size_t)rod * NE;
    float*       out_g = out       + (size_t)rod * (NE * 15);

    float epx = 0.f, epy = 0.f, epz = 0.f, rlp = 0.f;   // carried previous edge / rest length

    for (int t = 0; t < NTILES; ++t) {
        // double-buffer: kick off next tile's DMA, wait for current tile (in-order per wave)
        if (t + 1 < NTILES) {
            const int bb = (t + 1) & 1;
            tdm_issue(ldsBase + (u32)((bb * BUFF + w * WBUF) * 4),
                      gWave + (u64)((t + 1) * TILE * 3 * 4));
            __builtin_amdgcn_s_wait_tensorcnt(1);
        } else {
            __builtin_amdgcn_s_wait_tensorcnt(0);
        }

        const float* buf = &lds[(t & 1) * BUFF + w * WBUF + lane * ROWF];
        float p0x = buf[0], p0y = buf[1], p0z = buf[2];

        for (int i4 = 0; i4 < TILE; i4 += 4) {
            const int e0 = t * TILE + i4;               // multiple of 4
            const float4 rl4 = *(const float4*)(rl_g + e0);
            const float4 mk4 = *(const float4*)(mk_g + e0);
            const float4 th4 = *(const float4*)(th_g + e0);
            const float rlv[4] = {rl4.x, rl4.y, rl4.z, rl4.w};
            const float mkv[4] = {mk4.x, mk4.y, mk4.z, mk4.w};
            const float thv[4] = {th4.x, th4.y, th4.z, th4.w};
            float og[60];

            #pragma unroll
            for (int j = 0; j < 4; ++j) {
                const int i = i4 + j;
                const float p1x = buf[(i + 1) * 3 + 0];
                const float p1y = buf[(i + 1) * 3 + 1];
                const float p1z = buf[(i + 1) * 3 + 2];
                const float ex = p1x - p0x, ey = p1y - p0y, ez = p1z - p0z;
                const float rl = rlv[j];

                float kbx = 0.f, kby = 0.f, kbz = 0.f;
                if (e0 + j > 0) {
                    // kb = 2*cross(e_prev,e)/ (rl_prev*rl + dot(e_prev,e))
                    const float dt = epx * ex + epy * ey + epz * ez;
                    const float cx = epy * ez - epz * ey;
                    const float cy = epz * ex - epx * ez;
                    const float cz = epx * ey - epy * ex;
                    const float s2 = 2.0f / (rlp * rl + dt);
                    kbx = cx * s2; kby = cy * s2; kbz = cz * s2;
                    const float mag  = kbx * kbx + kby * kby + kbz * kbz;
                    const float invc = 1.0f / (4.0f + mag);
                    const float sinPhi = sqrtf(mag * invc);
                    const float cosPhi = sqrtf(4.0f * invc);
                    const float qs = sinPhi / fmaxf(sqrtf(mag), 1e-12f);
                    const float qx = kbx * qs, qy = kby * qs, qz = kbz * qs;
                    // rot = u + 2*(w*cross(q,u) + cross(q,cross(q,u)))
                    const float uvx = qy * uz - qz * uy;
                    const float uvy = qz * ux - qx * uz;
                    const float uvz = qx * uy - qy * ux;
                    const float wvx = qy * uvz - qz * uvy;
                    const float wvy = qz * uvx - qx * uvz;
                    const float wvz = qx * uvy - qy * uvx;
                    const float rx = ux + 2.0f * (cosPhi * uvx + wvx);
                    const float ry = uy + 2.0f * (cosPhi * uvy + wvy);
                    const float rz = uz + 2.0f * (cosPhi * uvz + wvz);
                    const bool keep = (1.0f - cosPhi) <= 1e-6f;
                    ux = keep ? ux : rx;
                    uy = keep ? uy : ry;
                    uz = keep ? uz : rz;
                }
                // b_v = normalize(cross(e, u)); then mask both
                float bvx = ey * uz - ez * uy;
                float bvy = ez * ux - ex * uz;
                float bvz = ex * uy - ey * ux;
                const float invb = 1.0f / fmaxf(sqrtf(bvx * bvx + bvy * bvy + bvz * bvz), 1e-12f);
                const float m  = mkv[j];
                const float sb = invb * m;
                bvx *= sb; bvy *= sb; bvz *= sb;
                const float bux = ux * m, buy = uy * m, buz = uz * m;
                float sq, cq;
                __sincosf(thv[j], &sq, &cq);
                og[j * 15 +  0] = bux;  og[j * 15 +  1] = buy;  og[j * 15 +  2] = buz;
                og[j * 15 +  3] = bvx;  og[j * 15 +  4] = bvy;  og[j * 15 +  5] = bvz;
                og[j * 15 +  6] = kbx;  og[j * 15 +  7] = kby;  og[j * 15 +  8] = kbz;
                og[j * 15 +  9] = cq * bux + sq * bvx;
                og[j * 15 + 10] = cq * buy + sq * bvy;
                og[j * 15 + 11] = cq * buz + sq * bvz;
                og[j * 15 + 12] = cq * bvx - sq * bux;
                og[j * 15 + 13] = cq * bvy - sq * buy;
                og[j * 15 + 14] = cq * bvz - sq * buz;

                p0x = p1x; p0y = p1y; p0z = p1z;
                epx = ex;  epy = ey;  epz = ez;  rlp = rl;
            }

            // 60 floats per 4-edge group, base (e0*60 B) is a multiple of 240 -> 16B aligned
            float4* op = (float4*)(out_g + (size_t)e0 * 15);
            #pragma unroll
            for (int q = 0; q < 15; ++q)
                op[q] = make_float4(og[4 * q + 0], og[4 * q + 1], og[4 * q + 2], og[4 * q + 3]);
        }
    }
}

extern "C" void kernel_launch(void* const* d_in, const int* in_sizes, int n_in,
                              void* d_out, int out_size, void* d_ws, size_t ws_size,
                              hipStream_t stream) {
    (void)in_sizes; (void)n_in; (void)out_size; (void)d_ws; (void)ws_size;
    const float* verts       = (const float*)d_in[0];
    const float* init_direct = (const float*)d_in[1];
    const float* m_theta     = (const float*)d_in[2];
    const float* edge_mask   = (const float*)d_in[3];
    const float* restEdgeL   = (const float*)d_in[4];
    float* out = (float*)d_out;

    dim3 grid(B_TOT / BLOCK);
    dim3 block(BLOCK);
    deft_rod_frames<<<grid, block, 0, stream>>>(verts, init_direct, m_theta,
                                                edge_mask, restEdgeL, out);
}